// HopfActivRadius_80221399154901
// MI455X (gfx1250) — compile-verified
//
#include <hip/hip_runtime.h>

// Elementwise Hopf-radius map:  r = r0 * exp(mu*T1) * rsqrt(1 + r0^2 * g)
//   g = expm1(2*mu*T1)/mu   (series for small mu)
// 8192x8192 fp32 -> memory bound (768 MB / 23.3 TB/s ~= 33 us floor).
// Strategy: b128 NT global loads/stores + v_exp_f32 / v_rcp_f32 / v_rsq_f32.

typedef float v4f __attribute__((ext_vector_type(4)));

namespace {
constexpr float kT1      = 0.19634954084936207f;   // 2*pi/32
constexpr float kTwoT1   = 0.39269908169872414f;   // 2*T1
constexpr float kT1Log2e = 0.28327252154716094f;   // T1 * log2(e)
}

__device__ __forceinline__ float hopf_radius(float r0, float mu) {
    // x = 2*mu*T1
    float x = kTwoT1 * mu;
    // E = exp(mu*T1) = exp2(mu * T1 * log2(e))  -> single v_exp_f32
    float E = __builtin_amdgcn_exp2f(kT1Log2e * mu);

    // Large-|x| path: expm1(2*mu*T1) = E*E - 1 (exact fma), then / mu via v_rcp_f32.
    float em    = __builtin_fmaf(E, E, -1.0f);
    float g_big = em * __builtin_amdgcn_rcpf(mu);

    // Small-|x| path (|x| < 1/16): g = 2*T1 * (1 + x/2 + x^2/6 + x^3/24 + x^4/120)
    float p = __builtin_fmaf(x, 1.0f / 120.0f, 1.0f / 24.0f);
    p       = __builtin_fmaf(x, p, 1.0f / 6.0f);
    p       = __builtin_fmaf(x, p, 0.5f);
    p       = __builtin_fmaf(x, p, 1.0f);
    float g_small = kTwoT1 * p;

    float g   = (__builtin_fabsf(x) < 0.0625f) ? g_small : g_big;  // v_cndmask, no divergence
    float den = __builtin_fmaf(r0 * r0, g, 1.0f);                  // >= 1, rsqrt safe
    return r0 * E * __builtin_amdgcn_rsqf(den);                    // v_rsq_f32
}

__global__ __launch_bounds__(256)
void hopf_v4_kernel(const v4f* __restrict__ z0,
                    const v4f* __restrict__ mu,
                    v4f* __restrict__ out,
                    unsigned int n4) {
    unsigned int i = blockIdx.x * 256u + threadIdx.x;
    if (i >= n4) return;

    // Streamed once, working set >> L2: non-temporal b128 loads/stores.
    v4f a = __builtin_nontemporal_load(z0 + i);
    v4f b = __builtin_nontemporal_load(mu + i);

    v4f r;
    r.x = hopf_radius(a.x, b.x);
    r.y = hopf_radius(a.y, b.y);
    r.z = hopf_radius(a.z, b.z);
    r.w = hopf_radius(a.w, b.w);

    __builtin_nontemporal_store(r, out + i);
}

__global__ __launch_bounds__(64)
void hopf_tail_kernel(const float* __restrict__ z0,
                      const float* __restrict__ mu,
                      float* __restrict__ out,
                      unsigned int n) {
    unsigned int i = threadIdx.x;
    if (i >= n) return;
    float r = hopf_radius(z0[i], mu[i]);
    __builtin_nontemporal_store(r, out + i);
}

extern "C" void kernel_launch(void* const* d_in, const int* in_sizes, int n_in,
                              void* d_out, int out_size, void* d_ws, size_t ws_size,
                              hipStream_t stream) {
    const float* z0  = (const float*)d_in[0];
    const float* mu  = (const float*)d_in[1];
    float*       out = (float*)d_out;

    const long long n   = (long long)out_size;       // 8192*8192 = 67,108,864
    const unsigned  n4  = (unsigned)(n >> 2);        // 16,777,216 float4s
    const unsigned  rem = (unsigned)(n & 3);         // 0 for this shape

    if (n4) {
        const unsigned blocks = (n4 + 255u) / 256u;  // 65,536 blocks x 8 wave32
        hopf_v4_kernel<<<blocks, 256, 0, stream>>>((const v4f*)z0, (const v4f*)mu,
                                                   (v4f*)out, n4);
    }
    if (rem) {
        const long long base = (long long)n4 * 4ll;
        hopf_tail_kernel<<<1, 64, 0, stream>>>(z0 + base, mu + base, out + base, rem);
    }
}